// MCPBRNN_Generic_PETconstraint_Two_VariantOutputGate_BYPASSM0_10196252360851
// MI455X (gfx1250) — compile-verified
//
#include <hip/hip_runtime.h>
#include <math.h>

namespace {

constexpr int T_     = 524288;
constexpr int SPIN_  = 365;
constexpr int TRAIN_ = 262144;
constexpr float ML_  = 2.9086f;
constexpr float SL_  = 1.898f;
constexpr int   NW_  = 512;   // reduction waves (1 wave per block)

typedef float v2f __attribute__((ext_vector_type(2)));
typedef float v8f __attribute__((ext_vector_type(8)));

struct ScalarPtrs {
  const float* p_mean;  const float* p_std;
  const float* w_r_yom; const float* w_r_gw; const float* w_r_ylm; const float* w_r_yfm;
  const float* w_b1_yom; const float* w_b1_gw; const float* w_b2_ylm; const float* theltaC;
  const float* b0_yom;  const float* b0_gw;   const float* b0_ylm;
};

struct Gates {
  float oo1, oogw1, ol1, expC, mo, inv_so, w1, wgw, wl, b1, bgw, bl;
};

__device__ __forceinline__ float sigm(float x) {
  return 1.0f / (1.0f + __expf(-x));
}

// SGPR broadcast of a float from a compile-time-constant lane.
__device__ __forceinline__ float bcast_lane(float v, int lane) {
  return __int_as_float(__builtin_amdgcn_readlane(__float_as_int(v), lane));
}

__device__ __forceinline__ Gates load_gates(const ScalarPtrs s) {
  Gates g;
  float e1 = __expf(s.w_r_yom[0]);
  float e2 = __expf(s.w_r_gw[0]);
  float e3 = __expf(s.w_r_ylm[0]);
  float e4 = __expf(s.w_r_yfm[0]);
  float inv = 1.0f / (e1 + e2 + e3 + e4);
  g.oo1 = e1 * inv; g.oogw1 = e2 * inv; g.ol1 = e3 * inv;
  g.expC   = __expf(s.theltaC[0]);
  g.mo     = s.p_mean[0];
  g.inv_so = 1.0f / s.p_std[0];
  g.w1  = s.w_b1_yom[0]; g.wgw = s.w_b1_gw[0]; g.wl = s.w_b2_ylm[0];
  g.b1  = s.b0_yom[0];   g.bgw = s.b0_gw[0];   g.bl = s.b0_ylm[0];
  return g;
}

// --------------------------------------------------------------------------
// Phase 1: moment reduction for obsstd via V_WMMA_F32_16X16X4_F32.
// A holds 64 data values (16x4 tile, one per element), B = all-ones 4x16.
// C[i][j] accumulates rowsum_i (identical across columns j), so the grand
// total is s(lane) + s(lane^16) where s = sum of the 8 C VGPRs.
// Tail handling via clamped index + select (EXEC stays all-ones — required
// around WMMA — and no per-iteration branches).
// --------------------------------------------------------------------------
__global__ void reduce_kernel(const float* __restrict__ y,
                              float* __restrict__ psum, float* __restrict__ psq) {
  const int wave = blockIdx.x;
  const int lane = threadIdx.x;
  const int n    = TRAIN_ - SPIN_;

  v8f cs = {};   // sum accumulator (16x16 f32 tile)
  v8f cq = {};   // sum-of-squares accumulator
  v2f ones; ones[0] = 1.0f; ones[1] = 1.0f;

  for (int base = wave * 64; base < n; base += NW_ * 64) {
    int idx  = SPIN_ + base + lane * 2;
    int idxc = (idx < TRAIN_ - 1) ? idx : (TRAIN_ - 2);  // always in-range
    float v0 = y[idxc];
    float v1 = y[idxc + 1];
    float e0 = (idx     < TRAIN_) ? v0 : 0.0f;
    float e1 = (idx + 1 < TRAIN_) ? v1 : 0.0f;
    v2f a;  a[0]  = e0;      a[1]  = e1;
    v2f a2; a2[0] = e0 * e0; a2[1] = e1 * e1;
    // D = A*B + C  (B = ones -> per-row running sums in C)
    cs = __builtin_amdgcn_wmma_f32_16x16x4_f32(false, a,  false, ones, (short)0, cs, false, false);
    cq = __builtin_amdgcn_wmma_f32_16x16x4_f32(false, a2, false, ones, (short)0, cq, false, false);
  }

  float s = cs[0]+cs[1]+cs[2]+cs[3]+cs[4]+cs[5]+cs[6]+cs[7];
  float q = cq[0]+cq[1]+cq[2]+cq[3]+cq[4]+cq[5]+cq[6]+cq[7];
  float s2 = s + __shfl(s, lane ^ 16);
  float q2 = q + __shfl(q, lane ^ 16);
  if (lane == 0) { psum[wave] = s2; psq[wave] = q2; }
}

// Deterministic fixed-order combine in double precision; ddof = 1.
__global__ void std_final_kernel(const float* __restrict__ psum,
                                 const float* __restrict__ psq,
                                 float* __restrict__ std_ws) {
  if (threadIdx.x == 0 && blockIdx.x == 0) {
    double s = 0.0, q = 0.0;
    for (int i = 0; i < NW_; ++i) { s += (double)psum[i]; q += (double)psq[i]; }
    double n   = (double)(TRAIN_ - SPIN_);
    double var = (q - s * s / n) / (n - 1.0);
    std_ws[0]  = (float)sqrt(var);
  }
}

// --------------------------------------------------------------------------
// Phase 2: the sequential recurrence. One wave32. Every 32 steps: one
// coalesced float2 load (software-pipelined) + per-lane ol precompute; the
// inner 32 steps are pure ALU. Per-step broadcasts use v_readlane_b32
// (compile-time lane after unroll) into SGPRs — off the carry critical path,
// no DS-pipeline latency. Stores only c0[t], one coalesced store / 32 steps.
// --------------------------------------------------------------------------
__global__ void scan_kernel(const float* __restrict__ x,
                            float* __restrict__ c_ws, ScalarPtrs sp) {
  const Gates g = load_gates(sp);
  const int lane = threadIdx.x;
  const float2* xp = reinterpret_cast<const float2*>(x);

  float c = 0.0f;                       // c_init
  float2 xv = xp[lane];                 // prefetch first chunk
  for (int base = 0; base < T_; base += 32) {
    float2 xnext;
    if (base + 32 < T_) xnext = xp[base + 32 + lane];
    else { xnext.x = 0.0f; xnext.y = 0.0f; }

    float u1l = xv.x, u2l = xv.y;
    // ol depends only on u2: compute 32 of them in parallel, off critical path
    float oll = g.ol1 * sigm(g.bl + (u2l - ML_) * (1.0f / SL_) * g.wl);

    float crec = 0.0f;
    #pragma unroll
    for (int k = 0; k < 32; ++k) {
      float u1 = bcast_lane(u1l, k);    // v_readlane_b32 -> SGPR
      float u2 = bcast_lane(u2l, k);
      float ol = bcast_lane(oll, k);
      crec = (lane == k) ? c : crec;    // record c0 entering step base+k

      float a    = (c - g.mo) * g.inv_so;
      float oo   = g.oo1   * sigm(g.b1  + a * g.w1);
      float oogw = g.oogw1 * sigm(g.bgw + a * g.wgw);
      float px   = fmaxf(u1 + c - g.expC, 0.0f);
      float olc  = (c > 0.0f) ? (ol - fmaxf(ol - u2 / c, 0.0f)) : ol;
      float f    = 1.0f - oo - oogw - olc;
      c = f * c + (u1 - px);
    }
    c_ws[base + lane] = crec;           // coalesced
  }
}

// --------------------------------------------------------------------------
// Phase 3: fully parallel recompute of all outputs from c0[t].
// Output layout (flat, return order):
//   [0..12)T : h_n, c_n, l_n, lc_n, bp_n, gw_n, g_ib, g_oo, g_ol, g_olc, g_f, g_oogw
//   [12T..14T): h_nout (T x 2 row-major: h_n[t], obsstd)
//   [14T..15T): obs
// --------------------------------------------------------------------------
__global__ void outputs_kernel(const float* __restrict__ x,
                               const float* __restrict__ c_ws,
                               const float* __restrict__ std_ws,
                               float* __restrict__ out, ScalarPtrs sp) {
  int t = blockIdx.x * blockDim.x + threadIdx.x;
  if (t >= T_) return;
  const Gates g = load_gates(sp);

  float u1 = x[2 * t], u2 = x[2 * t + 1];
  float c0 = c_ws[t];
  float obsstd = std_ws[0];

  float a    = (c0 - g.mo) * g.inv_so;
  float oo   = g.oo1   * sigm(g.b1  + a * g.w1);
  float oogw = g.oogw1 * sigm(g.bgw + a * g.wgw);
  float ol   = g.ol1   * sigm(g.bl + (u2 - ML_) * (1.0f / SL_) * g.wl);
  float px   = fmaxf(u1 + c0 - g.expC, 0.0f);
  float ib   = (u1 > 0.0f) ? (px / u1) : 0.0f;
  float olc  = (c0 > 0.0f) ? (ol - fmaxf(ol - u2 / c0, 0.0f)) : ol;
  float f    = 1.0f - oo - oogw - olc;
  float h    = oo * c0 + px;

  out[0  * T_ + t] = h;
  out[1  * T_ + t] = c0;
  out[2  * T_ + t] = ol * c0;
  out[3  * T_ + t] = olc * c0;
  out[4  * T_ + t] = px;
  out[5  * T_ + t] = oogw * c0;
  out[6  * T_ + t] = ib;
  out[7  * T_ + t] = oo;
  out[8  * T_ + t] = ol;
  out[9  * T_ + t] = olc;
  out[10 * T_ + t] = f;
  out[11 * T_ + t] = oogw;
  out[12 * T_ + 2 * t]     = h;
  out[12 * T_ + 2 * t + 1] = obsstd;
  out[14 * T_ + t]         = obsstd;
}

} // namespace

extern "C" void kernel_launch(void* const* d_in, const int* in_sizes, int n_in,
                              void* d_out, int out_size, void* d_ws, size_t ws_size,
                              hipStream_t stream) {
  const float* x     = (const float*)d_in[0];
  // d_in[1] = epoch, d_in[2] = time_lag (unused by the forward pass)
  const float* y_obs = (const float*)d_in[3];

  ScalarPtrs sp;
  sp.p_mean   = (const float*)d_in[4];
  sp.p_std    = (const float*)d_in[5];
  sp.w_r_yom  = (const float*)d_in[6];
  sp.w_r_gw   = (const float*)d_in[7];
  sp.w_r_ylm  = (const float*)d_in[8];
  sp.w_r_yfm  = (const float*)d_in[9];
  sp.w_b1_yom = (const float*)d_in[10];
  sp.w_b1_gw  = (const float*)d_in[11];
  sp.w_b2_ylm = (const float*)d_in[12];
  sp.theltaC  = (const float*)d_in[13];
  sp.b0_yom   = (const float*)d_in[14];
  sp.b0_gw    = (const float*)d_in[15];
  sp.b0_ylm   = (const float*)d_in[16];

  float* c_ws   = (float*)d_ws;        // T floats: carry sequence
  float* psum   = c_ws + T_;           // NW_ per-wave partial sums
  float* psq    = psum + NW_;          // NW_ per-wave partial sum-of-squares
  float* std_ws = psq + NW_;           // 1 float: obsstd
  float* out    = (float*)d_out;

  hipLaunchKernelGGL(reduce_kernel,    dim3(NW_),            dim3(32),  0, stream, y_obs, psum, psq);
  hipLaunchKernelGGL(std_final_kernel, dim3(1),              dim3(32),  0, stream, psum, psq, std_ws);
  hipLaunchKernelGGL(scan_kernel,      dim3(1),              dim3(32),  0, stream, x, c_ws, sp);
  hipLaunchKernelGGL(outputs_kernel,   dim3((T_ + 255)/256), dim3(256), 0, stream, x, c_ws, std_ws, out, sp);
}